// MLP_80376017977665
// MI455X (gfx1250) — compile-verified
//
#include <hip/hip_runtime.h>

// ---------------- problem constants ----------------
#define N_      100000
#define NFEAT_  500
#define M_      50000
#define KCL_    40          // clusters
#define NHID_   256
#define NCLS2_  1600        // 40*40

// ---------------- WMMA types ----------------
typedef __attribute__((ext_vector_type(2))) float v2f;
typedef __attribute__((ext_vector_type(8))) float v8f;

// ---------------- tiling ----------------
#define BM 128
#define BN 64
#define BK 64
#define LDA 68   // padded LDS row (floats), 16B aligned, spreads banks
#define LDB 68

// =====================================================================
// zero fill
// =====================================================================
__global__ void zero_f32(float* p, int n) {
    int i = blockIdx.x * 256 + threadIdx.x;
    if (i < n) p[i] = 0.0f;
}

// =====================================================================
// assign[m] = argmax(cluster_id[m, :40])
// =====================================================================
__global__ void compute_assign(const float* __restrict__ cluster_id,
                               int* __restrict__ assign, int Mv) {
    int m = blockIdx.x * 256 + threadIdx.x;
    if (m >= Mv) return;
    const float* row = cluster_id + (size_t)m * KCL_;
    int best = 0;
    float bv = row[0];
    #pragma unroll 8
    for (int j = 1; j < KCL_; ++j) {
        float v = row[j];
        if (v > bv) { bv = v; best = j; }
    }
    assign[m] = best;
}

// =====================================================================
// GEMM1: hg[m, n] = relu( sum_k x[cidx[m], k] * W1[n, k] + b1[n] )
//   M=50000 rows (gathered), K=500 (padded to 512), N=256
//   grid = (NHID/64, ceil(M/128)), block = 256 (8 wave32)
// =====================================================================
__global__ __launch_bounds__(256)
void fc1_gemm(const float* __restrict__ x,
              const int* __restrict__ cidx,
              const float* __restrict__ W1,
              const float* __restrict__ b1,
              float* __restrict__ hg, int Mv) {
    __shared__ float ldsA[BM * LDA];
    __shared__ float ldsBt[BN * LDB];

    const int tid  = threadIdx.x;
    const int lane = tid & 31;
    const int wv   = tid >> 5;
    const int lr   = lane & 15;
    const int kq   = (lane >> 4) << 1;          // 0 or 2
    const int rowBase = blockIdx.y * BM;
    const int colBase = blockIdx.x * BN;

    v8f acc[4] = {};

    for (int kb = 0; kb < 512; kb += BK) {
        // stage A tile (gathered x rows), zero-pad k >= 500 and rows >= M
        #pragma unroll
        for (int i = 0; i < 8; ++i) {
            int idx = tid + i * 256;
            int row = idx >> 4, j = idx & 15;
            int m = rowBase + row;
            int k = kb + j * 4;
            float4 v = {0.f, 0.f, 0.f, 0.f};
            if (m < Mv && k < NFEAT_) {
                int ci = cidx[m];
                v = *(const float4*)(x + (size_t)ci * NFEAT_ + k);
            }
            *(float4*)(ldsA + row * LDA + j * 4) = v;
        }
        // stage Bt tile: ldsBt[n][k] = W1[colBase+n, kb+k]
        #pragma unroll
        for (int i = 0; i < 4; ++i) {
            int idx = tid + i * 256;
            int n = idx >> 4, j = idx & 15;
            int k = kb + j * 4;
            float4 v = {0.f, 0.f, 0.f, 0.f};
            if (k < NFEAT_)
                v = *(const float4*)(W1 + (size_t)(colBase + n) * NFEAT_ + k);
            *(float4*)(ldsBt + n * LDB + j * 4) = v;
        }
        __syncthreads();

        #pragma unroll
        for (int k0 = 0; k0 < BK; k0 += 4) {
            v2f a = *(const v2f*)(ldsA + (wv * 16 + lr) * LDA + k0 + kq);
            #pragma unroll
            for (int ct = 0; ct < 4; ++ct) {
                v2f b = *(const v2f*)(ldsBt + (ct * 16 + lr) * LDB + k0 + kq);
                acc[ct] = __builtin_amdgcn_wmma_f32_16x16x4_f32(
                    false, a, false, b, (short)0, acc[ct], false, false);
            }
        }
        __syncthreads();
    }

    // epilogue: bias + relu
    #pragma unroll
    for (int r = 0; r < 8; ++r) {
        int grow = rowBase + wv * 16 + ((lane >> 4) << 3) + r;
        if (grow < Mv) {
            #pragma unroll
            for (int ct = 0; ct < 4; ++ct) {
                int gcol = colBase + ct * 16 + lr;
                float v = acc[ct][r] + b1[gcol];
                hg[(size_t)grow * NHID_ + gcol] = v > 0.f ? v : 0.f;
            }
        }
    }
}

// =====================================================================
// segment sums: gsum[c, :] += hg[m, :] for assign[m]==c ; gcount[c] += 1
//   Race-free per block: thread t exclusively owns column t of LDS accum.
//   grid = ceil(M/256), block = 256
// =====================================================================
__global__ __launch_bounds__(256)
void seg_sum(const float* __restrict__ hg,
             const int* __restrict__ assign,
             float* __restrict__ gsum,
             float* __restrict__ gcount, int Mv) {
    __shared__ float acc[KCL_ * NHID_];
    __shared__ int   asn[256];
    __shared__ float cnt[KCL_];

    const int t = threadIdx.x;
    for (int i = t; i < KCL_ * NHID_; i += 256) acc[i] = 0.f;
    if (t < KCL_) cnt[t] = 0.f;

    const int mBase = blockIdx.x * 256;
    int m = mBase + t;
    asn[t] = (m < Mv) ? assign[m] : -1;
    __syncthreads();

    for (int r = 0; r < 256; ++r) {
        int c = asn[r];
        if (c >= 0) {
            acc[c * NHID_ + t] += hg[(size_t)(mBase + r) * NHID_ + t];
            if (t == 0) cnt[c] += 1.f;
        }
    }
    __syncthreads();

    for (int i = t; i < KCL_ * NHID_; i += 256) atomicAdd(&gsum[i], acc[i]);
    if (t < KCL_) atomicAdd(&gcount[t], cnt[t]);
}

// =====================================================================
// cf = gsum / gcount ; cfa[c,n] = cf[c]·W2[n, 0:256]   + b2[n]
//                      cfb[c,n] = cf[c]·W2[n, 256:512] + b2[n]
//   grid = 40, block = 256
// =====================================================================
__global__ __launch_bounds__(256)
void cluster_fc(const float* __restrict__ gsum,
                const float* __restrict__ gcount,
                const float* __restrict__ W2,
                const float* __restrict__ b2,
                float* __restrict__ cfa,
                float* __restrict__ cfb) {
    __shared__ float cf[NHID_];
    const int c = blockIdx.x, t = threadIdx.x;
    cf[t] = gsum[c * NHID_ + t] / gcount[c];
    __syncthreads();
    for (int n = t; n < NCLS2_; n += 256) {
        const float* wr = W2 + (size_t)n * (2 * NHID_);
        float sa = 0.f, sb = 0.f;
        #pragma unroll 8
        for (int k = 0; k < NHID_; ++k) {
            float ck = cf[k];
            sa = fmaf(ck, wr[k], sa);
            sb = fmaf(ck, wr[NHID_ + k], sb);
        }
        cfa[c * NCLS2_ + n] = sa + b2[n];
        cfb[c * NCLS2_ + n] = sb + b2[n];
    }
}

// =====================================================================
// GEMM2: out[outRowOff+m, n] = hg[m,:]·W2[n, kOff:kOff+256] + addMat[assign[m], n]
//   grid = (1600/64, ceil(M/128)), block = 256
// =====================================================================
__global__ __launch_bounds__(256)
void fc2_gemm(const float* __restrict__ hg,
              const float* __restrict__ W2, int kOff,
              const float* __restrict__ addMat,
              const int* __restrict__ assign,
              float* __restrict__ out, int outRowOff, int Mv) {
    __shared__ float ldsA[BM * LDA];
    __shared__ float ldsBt[BN * LDB];

    const int tid  = threadIdx.x;
    const int lane = tid & 31;
    const int wv   = tid >> 5;
    const int lr   = lane & 15;
    const int kq   = (lane >> 4) << 1;
    const int rowBase = blockIdx.y * BM;
    const int colBase = blockIdx.x * BN;

    v8f acc[4] = {};

    for (int kb = 0; kb < NHID_; kb += BK) {
        #pragma unroll
        for (int i = 0; i < 8; ++i) {
            int idx = tid + i * 256;
            int row = idx >> 4, j = idx & 15;
            int m = rowBase + row;
            float4 v = {0.f, 0.f, 0.f, 0.f};
            if (m < Mv)
                v = *(const float4*)(hg + (size_t)m * NHID_ + kb + j * 4);
            *(float4*)(ldsA + row * LDA + j * 4) = v;
        }
        #pragma unroll
        for (int i = 0; i < 4; ++i) {
            int idx = tid + i * 256;
            int n = idx >> 4, j = idx & 15;
            float4 v = *(const float4*)(W2 + (size_t)(colBase + n) * (2 * NHID_)
                                           + kOff + kb + j * 4);
            *(float4*)(ldsBt + n * LDB + j * 4) = v;
        }
        __syncthreads();

        #pragma unroll
        for (int k0 = 0; k0 < BK; k0 += 4) {
            v2f a = *(const v2f*)(ldsA + (wv * 16 + lr) * LDA + k0 + kq);
            #pragma unroll
            for (int ct = 0; ct < 4; ++ct) {
                v2f b = *(const v2f*)(ldsBt + (ct * 16 + lr) * LDB + k0 + kq);
                acc[ct] = __builtin_amdgcn_wmma_f32_16x16x4_f32(
                    false, a, false, b, (short)0, acc[ct], false, false);
            }
        }
        __syncthreads();
    }

    // epilogue: add gathered rank-40 term (already contains b2)
    #pragma unroll
    for (int r = 0; r < 8; ++r) {
        int grow = rowBase + wv * 16 + ((lane >> 4) << 3) + r;
        if (grow < Mv) {
            int c = assign[grow];
            const float* av = addMat + (size_t)c * NCLS2_;
            float* orow = out + (size_t)(outRowOff + grow) * NCLS2_;
            #pragma unroll
            for (int ct = 0; ct < 4; ++ct) {
                int gcol = colBase + ct * 16 + lr;
                orow[gcol] = acc[ct][r] + av[gcol];
            }
        }
    }
}

// =====================================================================
extern "C" void kernel_launch(void* const* d_in, const int* in_sizes, int n_in,
                              void* d_out, int out_size, void* d_ws, size_t ws_size,
                              hipStream_t stream) {
    const float* x          = (const float*)d_in[0];
    /* d_in[1] = adj, unused */
    const float* cluster_id = (const float*)d_in[2];
    const int*   cidx       = (const int*)d_in[3];
    const float* W1         = (const float*)d_in[4];
    const float* b1         = (const float*)d_in[5];
    const float* W2         = (const float*)d_in[6];
    const float* b2         = (const float*)d_in[7];
    float* out = (float*)d_out;

    char* ws = (char*)d_ws;
    size_t off = 0;
    float* hg     = (float*)(ws + off); off += (size_t)M_ * NHID_ * 4;   // 51.2 MB
    float* gsum   = (float*)(ws + off); off += (size_t)KCL_ * NHID_ * 4; // 40 KB
    float* gcount = (float*)(ws + off); off += 64 * 4;                   // padded
    float* cfa    = (float*)(ws + off); off += (size_t)KCL_ * NCLS2_ * 4;
    float* cfb    = (float*)(ws + off); off += (size_t)KCL_ * NCLS2_ * 4;
    int*   assign = (int*)  (ws + off); off += (size_t)M_ * 4;

    const int nZero = KCL_ * NHID_ + 64;  // gsum + gcount (contiguous)
    zero_f32<<<(nZero + 255) / 256, 256, 0, stream>>>(gsum, nZero);

    compute_assign<<<(M_ + 255) / 256, 256, 0, stream>>>(cluster_id, assign, M_);

    dim3 g1(NHID_ / BN, (M_ + BM - 1) / BM);          // (4, 391)
    fc1_gemm<<<g1, 256, 0, stream>>>(x, cidx, W1, b1, hg, M_);

    seg_sum<<<(M_ + 255) / 256, 256, 0, stream>>>(hg, assign, gsum, gcount, M_);

    cluster_fc<<<KCL_, 256, 0, stream>>>(gsum, gcount, W2, b2, cfa, cfb);

    dim3 g2(NCLS2_ / BN, (M_ + BM - 1) / BM);         // (25, 391)
    // top half:    hg @ W2a.T + (cf@W2b.T + b2)[assign]
    fc2_gemm<<<g2, 256, 0, stream>>>(hg, W2, 0,     cfb, assign, out, 0,  M_);
    // bottom half: hg @ W2b.T + (cf@W2a.T + b2)[assign]
    fc2_gemm<<<g2, 256, 0, stream>>>(hg, W2, NHID_, cfa, assign, out, M_, M_);
}